// TorchCorr_67370857005292
// MI455X (gfx1250) — compile-verified
//
#include <hip/hip_runtime.h>

typedef __attribute__((ext_vector_type(2))) float v2f;
typedef __attribute__((ext_vector_type(8))) float v8f;

// Problem constants: N,C,HR,WR,HS,WS,S = 4,128,64,64,64,64,64
constexpr int N_  = 4;
constexpr int C_  = 128;
constexpr int HR_ = 64, WR_ = 64;
constexpr int HS_ = 64, WS_ = 64;
constexpr int S_  = 64;
constexpr int P_  = HR_ * WR_;   // 4096
constexpr int Q_  = HS_ * WS_;   // 4096

constexpr int PT      = 32;              // p-rows per WG = 2 WMMA M-tiles (B reused 2x)
constexpr int QC      = 2048;            // q-chunk resident in LDS
constexpr int NCHUNK  = Q_ / QC;         // 2
constexpr int QS      = QC + 4;          // padded LDS row stride (bank rotation)
constexpr int NWAVE   = 8;
constexpr int NTHREADS= NWAVE * 32;
constexpr int KSTEPS  = C_ / 4;          // 32 wmma k-steps
constexpr int TPW     = (QC / 16) / NWAVE;       // 16 q-tiles per wave per chunk
constexpr int SPT     = PT * S_ / NTHREADS;      // 8 samples per thread

__global__ __launch_bounds__(NTHREADS)
void corr_fused_wmma2(const float* __restrict__ feat_ref,
                      const float* __restrict__ feat_src,
                      const float* __restrict__ grids,
                      const float* __restrict__ mask,
                      float* __restrict__ out)
{
    // 32 x 2052 fp32 corr chunk = 262,656 B of the WGP's 320 KB LDS
    __shared__ float lds_corr[PT * QS];

    const int bid = blockIdx.x;
    const int n   = bid >> 7;          // / 128
    const int pt  = bid & 127;
    const int p0  = pt * PT;

    const int tid  = threadIdx.x;
    const int wave = tid >> 5;
    const int lane = tid & 31;
    const int half = lane >> 4;
    const int l16  = lane & 15;
    const int kb   = half * 2;

    const float* frn = feat_ref + (size_t)n * C_ * P_;
    const float* fsn = feat_src + (size_t)n * C_ * Q_;

    // per-thread sampling partials, accumulated across q-chunks
    float o[SPT], ms[SPT];
#pragma unroll
    for (int i = 0; i < SPT; ++i) { o[i] = 0.f; ms[i] = 0.f; }

    const int pi = tid & 31;   // consecutive threads -> consecutive p (coalesced)
    const int sb = tid >> 5;   // 0..7, sample s = sb + i*8

#pragma unroll 1
    for (int chunk = 0; chunk < NCHUNK; ++chunk) {
        // ---- A tiles (fr^T, two 16x128 M-tiles) in WMMA A layout ----
        // reloaded per chunk (L2-hot) so the 128 VGPRs aren't live during sampling
        v2f a0[KSTEPS], a1[KSTEPS];
#pragma unroll
        for (int ks = 0; ks < KSTEPS; ++ks) {
            const int c = ks * 4 + kb;
            const float* r0 = frn + (size_t)c * P_ + p0 + l16;
            const float* r1 = r0 + P_;
            a0[ks].x = r0[0];  a0[ks].y = r1[0];    // rows p0..p0+15
            a1[ks].x = r0[16]; a1[ks].y = r1[16];   // rows p0+16..p0+31
        }

        // ---- GEMM: corr[32 x 2048] chunk; every B load feeds 2 WMMAs ----
#pragma unroll 1
        for (int g = 0; g < TPW / 2; ++g) {
            const int tb  = wave * TPW + g * 2;
            const int col = tb * 16 + l16;        // column within chunk
            const int q0  = chunk * QC + col;     // global q
            v8f acc00 = {}, acc01 = {}, acc10 = {}, acc11 = {}; // [mtile][qtile]
#pragma unroll
            for (int ks = 0; ks < KSTEPS; ++ks) {
                const int c = ks * 4 + kb;
                const float* row0 = fsn + (size_t)c * Q_;
                const float* row1 = row0 + Q_;
                v2f b0, b1;
                b0.x = row0[q0];      b0.y = row1[q0];
                b1.x = row0[q0 + 16]; b1.y = row1[q0 + 16];
                acc00 = __builtin_amdgcn_wmma_f32_16x16x4_f32(false, a0[ks], false, b0,
                                                              (short)0, acc00, false, false);
                acc10 = __builtin_amdgcn_wmma_f32_16x16x4_f32(false, a1[ks], false, b0,
                                                              (short)0, acc10, false, false);
                acc01 = __builtin_amdgcn_wmma_f32_16x16x4_f32(false, a0[ks], false, b1,
                                                              (short)0, acc01, false, false);
                acc11 = __builtin_amdgcn_wmma_f32_16x16x4_f32(false, a1[ks], false, b1,
                                                              (short)0, acc11, false, false);
            }
            // D layout: VGPR r -> M = r + 8*half, N = lane&15
#pragma unroll
            for (int r = 0; r < 8; ++r) {
                const int row = r + 8 * half;
                lds_corr[row * QS + col]             = acc00[r];
                lds_corr[row * QS + col + 16]        = acc01[r];
                lds_corr[(row + 16) * QS + col]      = acc10[r];
                lds_corr[(row + 16) * QS + col + 16] = acc11[r];
            }
        }
        __syncthreads();

        // ---- partial bilinear sampling against the resident q-chunk ----
        const float* crow = &lds_corr[pi * QS];
        const int qlo = chunk * QC;
#pragma unroll
        for (int i = 0; i < SPT; ++i) {
            const int s = sb + i * 8;
            const int p = p0 + pi;
            const size_t gbase = ((size_t)(n * S_ + s) * 2) * P_ + p;
            const float gx = grids[gbase];
            const float gy = grids[gbase + P_];

            const float ixf = gx - 0.5f, iyf = gy - 0.5f;
            const float x0f = floorf(ixf), y0f = floorf(iyf);
            const float wx1 = ixf - x0f,  wy1 = iyf - y0f;
            const int x0i = (int)x0f, y0i = (int)y0f;
#pragma unroll
            for (int dx = 0; dx < 2; ++dx) {
                const float wx = dx ? wx1 : (1.f - wx1);
                const int   xi = x0i + dx;
                const int   xc = xi < 0 ? 0 : (xi > WS_ - 1 ? WS_ - 1 : xi);
                const bool  xin = (xi >= 0) & (xi < WS_);
#pragma unroll
                for (int dy = 0; dy < 2; ++dy) {
                    const float wy = dy ? wy1 : (1.f - wy1);
                    const int   yi = y0i + dy;
                    const int   yc = yi < 0 ? 0 : (yi > HS_ - 1 ? HS_ - 1 : yi);
                    const bool  in = xin & (yi >= 0) & (yi < HS_);
                    const float w  = wx * wy * (in ? 1.f : 0.f);
                    const int   li = yc * WS_ + xc - qlo;
                    if ((unsigned)li < (unsigned)QC) o[i] += w * crow[li];
                    if (chunk == 0) ms[i] += w;   // weights are chunk-independent
                }
            }
        }
        __syncthreads();   // before next chunk's GEMM overwrites LDS
    }

    // ---- finalize: coverage mask + user mask, write both outputs ----
#pragma unroll
    for (int i = 0; i < SPT; ++i) {
        const int s = sb + i * 8;
        const int p = p0 + pi;
        const float cm = (ms[i] < 0.9999f) ? 0.f : 1.f;
        const size_t oidx = (size_t)(n * S_ + s) * P_ + p;
        const float  mv   = mask[oidx];
        out[oidx]                        = o[i] * cm * mv;   // output 0: sampled corr
        out[(size_t)N_ * S_ * P_ + oidx] = cm * mv;          // output 1: corr_mask
    }
}

extern "C" void kernel_launch(void* const* d_in, const int* in_sizes, int n_in,
                              void* d_out, int out_size, void* d_ws, size_t ws_size,
                              hipStream_t stream) {
    (void)in_sizes; (void)n_in; (void)out_size; (void)d_ws; (void)ws_size;
    const float* feat_ref = (const float*)d_in[0];
    const float* feat_src = (const float*)d_in[1];
    const float* grids    = (const float*)d_in[2];
    const float* mask     = (const float*)d_in[3];
    float* out = (float*)d_out;

    dim3 grid(N_ * (P_ / PT));   // 4 * 128 = 512 workgroups
    dim3 block(NTHREADS);        // 8 wave32s
    hipLaunchKernelGGL(corr_fused_wmma2, grid, block, 0, stream,
                       feat_ref, feat_src, grids, mask, out);
}